// VanillaRNN_55508157334109
// MI455X (gfx1250) — compile-verified
//
#include <hip/hip_runtime.h>
#include <math.h>

#define VOCAB 50257
#define HID   2048
#define KSPLIT (HID / 8)   // 256 k-values per wave, 8 waves per block
#define PF_DIST 64         // prefetch distance in rows (4 loop bodies ahead)

typedef float v2f __attribute__((ext_vector_type(2)));
typedef float v8f __attribute__((ext_vector_type(8)));

#define WMMA_F32_4(A, B, C) \
  __builtin_amdgcn_wmma_f32_16x16x4_f32(false, (A), false, (B), (short)0, (C), false, false)

// ---------------------------------------------------------------------------
// Wave-level partial GEMV over k in [kbeg, kend) using V_WMMA_F32_16X16X4_F32.
//
// acc[n] = sum_{k=kbeg}^{kend-1} h[k] * W[k*ld + v0 + n],   n = 0..15
//
// All 16 rows of the A matrix hold identical copies of the activation slice
// h[k0..k0+3], so every row of D is the same GEMV partial:
//   A 16x4 f32 : lanes 0-15 -> {K=0,K=1} in v0/v1, lanes 16-31 -> {K=2,K=3}
//   B 4x16 f32 : lanes 0-15 -> rows K=0/K=1, lanes 16-31 -> rows K=2/K=3,
//                column = lane & 15
//   D 16x16 f32: c[0] of lane L holds the result for column v0 + (L&15)
//
// Depth-4 modulo-scheduled pipeline with fixed register stages (no rotation
// movs): each stage is reloaded right after the WMMA that consumes it, so 3
// stages (6 global + 3 ds loads) stay in flight across every WMMA.
// Prefetch uses its own lane mapping (lane L -> row kp + (L>>3), byte offset
// (L&7)*8) so ONE global_prefetch_b8 covers 4 consecutive rows of the 64 B
// tile span; 4 per body cover every row, PF_DIST rows ahead, clamped with a
// per-lane min (no EXEC manipulation -> EXEC stays all-1s for every WMMA).
// ---------------------------------------------------------------------------
__device__ __forceinline__ float gemv16_wmma_range(const float* __restrict__ W,
                                                   int ld,
                                                   const float* __restrict__ h,
                                                   int kbeg, int kend, int Ktot,
                                                   int v0, int vmax) {
  const int lane = threadIdx.x & 31;
  const int n    = lane & 15;
  const int kh   = (lane >> 4) << 1;       // 0 for lanes 0-15, 2 for 16-31
  int v = v0 + n;
  if (v > vmax) v = vmax;                  // clamp loads; keep EXEC uniform
  const float* Wp = W + (size_t)v;

  // prefetch lane mapping: 8 lanes x 8 B span one row, 4 rows per instruction
  const int prow = lane >> 3;              // 0..3
  int vp = v0 + ((lane & 7) << 1);
  if (vp > vmax) vp = vmax;
  const float* Pp = W + (size_t)vp;

#define LOAD_STAGE(As, Bs, kk)                                                 \
  do {                                                                         \
    const int k_ = (kk) + kh;                                                  \
    As.x = h[k_];                         /* LDS broadcast in half-wave */     \
    As.y = h[k_ + 1];                                                          \
    Bs.x = Wp[(size_t)k_ * (size_t)ld];                                        \
    Bs.y = Wp[(size_t)(k_ + 1) * (size_t)ld];                                  \
  } while (0)

#define PREFETCH_ROWS(kk)                                                      \
  do {                                                                         \
    int kp_ = (kk) + prow;                                                     \
    if (kp_ > Ktot - 1) kp_ = Ktot - 1;   /* v_min, no EXEC change */          \
    __builtin_prefetch(&Pp[(size_t)kp_ * (size_t)ld], 0, 1);                   \
  } while (0)

  v2f A0, B0, A1, B1, A2, B2, A3, B3;
  LOAD_STAGE(A0, B0, kbeg);
  LOAD_STAGE(A1, B1, kbeg + 4);
  LOAD_STAGE(A2, B2, kbeg + 8);
  LOAD_STAGE(A3, B3, kbeg + 12);

  v8f c = {};
  for (int k0 = kbeg; k0 <= kend - 32; k0 += 16) {
    PREFETCH_ROWS(k0 + PF_DIST);
    c = WMMA_F32_4(A0, B0, c);
    LOAD_STAGE(A0, B0, k0 + 16);

    PREFETCH_ROWS(k0 + PF_DIST + 4);
    c = WMMA_F32_4(A1, B1, c);
    LOAD_STAGE(A1, B1, k0 + 20);

    PREFETCH_ROWS(k0 + PF_DIST + 8);
    c = WMMA_F32_4(A2, B2, c);
    LOAD_STAGE(A2, B2, k0 + 24);

    PREFETCH_ROWS(k0 + PF_DIST + 12);
    c = WMMA_F32_4(A3, B3, c);
    LOAD_STAGE(A3, B3, k0 + 28);
  }
  // epilogue: consume the 4 in-flight stages
  c = WMMA_F32_4(A0, B0, c);
  c = WMMA_F32_4(A1, B1, c);
  c = WMMA_F32_4(A2, B2, c);
  c = WMMA_F32_4(A3, B3, c);

#undef LOAD_STAGE
#undef PREFETCH_ROWS
  return c[0];
}

// ---------------------------------------------------------------------------
// Kernel 1: h = tanh(IH[current] + prev @ HH + b_h)          (2048 outputs)
// One 16-column tile per block; 8 waves split K=2048; LDS reduction.
// 128 blocks x 8 waves over the 16 MB HH stream.
// ---------------------------------------------------------------------------
__global__ void rnn_hidden_kernel(const int*   __restrict__ cur_p,
                                  const float* __restrict__ prev,
                                  const float* __restrict__ IH,
                                  const float* __restrict__ HH,
                                  const float* __restrict__ bh,
                                  float* __restrict__ out_h,   // d_out + VOCAB
                                  float* __restrict__ ws_h) {  // d_ws
  __shared__ float sh[HID];
  __shared__ float part[8][16];
  for (int i = threadIdx.x; i < HID; i += blockDim.x) sh[i] = prev[i];
  __syncthreads();

  const int wave = threadIdx.x >> 5;
  const int lane = threadIdx.x & 31;
  const int j0   = blockIdx.x << 4;
  const int kbeg = wave * KSPLIT;

  const float acc = gemv16_wmma_range(HH, HID, sh, kbeg, kbeg + KSPLIT, HID,
                                      j0, HID - 1);

  if (lane < 16) part[wave][lane] = acc;
  __syncthreads();

  if (threadIdx.x < 16) {
    float s = 0.0f;
#pragma unroll
    for (int w = 0; w < 8; ++w) s += part[w][threadIdx.x];
    const int j   = j0 + threadIdx.x;
    const int cur = cur_p[0];
    const float hp = s + IH[(size_t)cur * HID + (size_t)j] + bh[j];
    const float hv = tanhf(hp);
    out_h[j] = hv;
    ws_h[j]  = hv;
  }
}

// ---------------------------------------------------------------------------
// Kernel 2: out = h @ HO + b_o                              (50257 outputs)
// One 16-column tile per block; 8 waves split K=2048; LDS reduction.
// 3142 blocks x 8 waves = ~25k waves streaming the 412 MB HO matrix:
// ~18 us floor at 23.3 TB/s, with enough waves/SIMD to cover HBM latency.
// ---------------------------------------------------------------------------
__global__ void rnn_output_kernel(const float* __restrict__ HO,
                                  const float* __restrict__ bo,
                                  const float* __restrict__ hact, // d_ws
                                  float* __restrict__ out) {      // d_out
  __shared__ float sh[HID];
  __shared__ float part[8][16];
  for (int i = threadIdx.x; i < HID; i += blockDim.x) sh[i] = hact[i];
  __syncthreads();

  const int wave = threadIdx.x >> 5;
  const int lane = threadIdx.x & 31;
  const int v0   = blockIdx.x << 4;
  const int kbeg = wave * KSPLIT;

  const float acc = gemv16_wmma_range(HO, VOCAB, sh, kbeg, kbeg + KSPLIT, HID,
                                      v0, VOCAB - 1);

  if (lane < 16) part[wave][lane] = acc;
  __syncthreads();

  if (threadIdx.x < 16) {
    float s = 0.0f;
#pragma unroll
    for (int w = 0; w < 8; ++w) s += part[w][threadIdx.x];
    const int v = v0 + threadIdx.x;
    if (v < VOCAB) out[v] = s + bo[v];
  }
}

// ---------------------------------------------------------------------------
extern "C" void kernel_launch(void* const* d_in, const int* in_sizes, int n_in,
                              void* d_out, int out_size, void* d_ws, size_t ws_size,
                              hipStream_t stream) {
  const int*   cur  = (const int*)  d_in[0];  // current token (1 elem)
  const float* prev = (const float*)d_in[1];  // [1, HID]
  const float* IH   = (const float*)d_in[2];  // [VOCAB, HID]
  const float* HH   = (const float*)d_in[3];  // [HID, HID]
  const float* HO   = (const float*)d_in[4];  // [HID, VOCAB]
  const float* bh   = (const float*)d_in[5];  // [1, HID]
  const float* bo   = (const float*)d_in[6];  // [1, VOCAB]

  float* out = (float*)d_out;                 // [VOCAB] output ++ [HID] h_new
  float* wsf = (float*)d_ws;                  // scratch: h_new for kernel 2

  const dim3 blk(256);                        // 8 wave32 waves

  rnn_hidden_kernel<<<HID / 16, blk, 0, stream>>>(cur, prev, IH, HH, bh,
                                                  out + VOCAB, wsf);

  const int tiles = (VOCAB + 15) / 16;        // 3142 column tiles
  rnn_output_kernel<<<tiles, blk, 0, stream>>>(HO, bo, wsf, out);
}